// LIFActivation_283467841909
// MI455X (gfx1250) — compile-verified
//
#include <hip/hip_runtime.h>
#include <hip/hip_bf16.h>
#include <stdint.h>

// ---------------------------------------------------------------------------
// Element-wise IF-neuron rate (T=4, subtractive reset) on MI455X (gfx1250).
// Pure HBM stream: 268 MB traffic -> ~11.5us floor @ 23.3 TB/s. No matrix
// structure, so no WMMA; the CDNA5 feature exercised is the async
// global->LDS pipeline (global_load_async_to_lds_b128 + s_wait_asynccnt)
// plus non-temporal 128-bit stores.
// ---------------------------------------------------------------------------

typedef float v4f __attribute__((ext_vector_type(4)));

#define BLOCK 256
#define MAX_GRID 4096
#define EPSF 1e-12f

// Issue a per-lane 16B async copy global->LDS. vdst = LDS byte address,
// vaddr = 64-bit global address. Tracked by ASYNCcnt.
__device__ __forceinline__ void async_load_b128(uint32_t lds_addr, const void* gaddr) {
    asm volatile("global_load_async_to_lds_b128 %0, %1, off"
                 :
                 : "v"(lds_addr), "v"(gaddr)
                 : "memory");
}

template <int N>
__device__ __forceinline__ void wait_async() {
#if __has_builtin(__builtin_amdgcn_s_wait_asynccnt)
    __builtin_amdgcn_s_wait_asynccnt(N);
#else
    asm volatile("s_wait_asynccnt %0" :: "n"(N) : "memory");
#endif
    asm volatile("" ::: "memory");  // keep LDS reads below the wait
}

// Bit-exact IF neuron: v<-v+u; s=H(v-1); v<-v-s; repeated T=4 times.
// (v >= 1.0f) == (v - 1.0f >= 0.0f) for all finite v (Sterbenz on [0.5,2],
// trivially equivalent outside), so this matches the reference exactly.
__device__ __forceinline__ float if_rate(float x, float safe_scale) {
    float u = fmaxf(x, 0.0f) / safe_scale;   // IEEE divide, matches jnp
    float v = 0.0f;
    float cnt = 0.0f;
#pragma unroll
    for (int t = 0; t < 4; ++t) {
        v += u;
        float sp = (v >= 1.0f) ? 1.0f : 0.0f;
        v -= sp;
        cnt += sp;
    }
    return (cnt * 0.25f) * safe_scale;       // (sum/4)*scale, *0.25 exact
}

__global__ __launch_bounds__(BLOCK) void if_rate_kernel(
    const float* __restrict__ x,
    const float* __restrict__ scale,
    float* __restrict__ out,
    size_t n) {
    __shared__ v4f tile[2][BLOCK];

    const float ss = fmaxf(scale[0], EPSF);  // uniform scalar load
    const unsigned tid = threadIdx.x;
    const size_t nvec   = n >> 2;            // # of float4
    const size_t ntiles = nvec / BLOCK;      // full 4KB tiles
    const size_t stride = gridDim.x;         // block-stride over tiles

    const v4f* __restrict__ xv = (const v4f*)x;
    v4f* __restrict__ ov = (v4f*)out;

    const uint32_t lds0 = (uint32_t)(uintptr_t)&tile[0][tid];
    const uint32_t lds1 = (uint32_t)(uintptr_t)&tile[1][tid];

    // -------- double-buffered async global->LDS pipeline over tiles --------
    size_t g = blockIdx.x;
    if (g < ntiles) {
        async_load_b128(lds0, xv + g * BLOCK + tid);
        unsigned buf = 0;
        while (g < ntiles) {
            const size_t gn = g + stride;
            const bool more = gn < ntiles;
            if (more) async_load_b128(buf ? lds0 : lds1, xv + gn * BLOCK + tid);
            if (more) wait_async<1>(); else wait_async<0>();

            v4f val = tile[buf][tid];        // ds_load_b128 (own slot, no barrier)
            v4f r;
            r.x = if_rate(val.x, ss);
            r.y = if_rate(val.y, ss);
            r.z = if_rate(val.z, ss);
            r.w = if_rate(val.w, ss);
            __builtin_nontemporal_store(r, ov + g * BLOCK + tid);  // b128 NT store

            g = gn;
            buf ^= 1u;
        }
    }

    // -------- tails (empty for 4096x8192, kept for generality) --------
    const size_t gid = (size_t)blockIdx.x * BLOCK + tid;
    const size_t nthreads = (size_t)gridDim.x * BLOCK;
    for (size_t j = ntiles * BLOCK + gid; j < nvec; j += nthreads) {
        v4f val = __builtin_nontemporal_load(xv + j);
        v4f r;
        r.x = if_rate(val.x, ss);
        r.y = if_rate(val.y, ss);
        r.z = if_rate(val.z, ss);
        r.w = if_rate(val.w, ss);
        __builtin_nontemporal_store(r, ov + j);
    }
    for (size_t j = (nvec << 2) + gid; j < n; j += nthreads) {
        out[j] = if_rate(x[j], ss);
    }
}

extern "C" void kernel_launch(void* const* d_in, const int* in_sizes, int n_in,
                              void* d_out, int out_size, void* d_ws, size_t ws_size,
                              hipStream_t stream) {
    (void)n_in; (void)d_ws; (void)ws_size; (void)out_size;
    const float* x     = (const float*)d_in[0];
    const float* scale = (const float*)d_in[1];
    float* out = (float*)d_out;

    const size_t n = (size_t)in_sizes[0];
    const size_t ntiles = (n >> 2) / BLOCK;
    int grid = (int)((ntiles == 0) ? 1 : (ntiles < MAX_GRID ? ntiles : MAX_GRID));

    if_rate_kernel<<<grid, BLOCK, 0, stream>>>(x, scale, out, n);
}